// CombinedLoss_77532749627835
// MI455X (gfx1250) — compile-verified
//
#include <hip/hip_runtime.h>

#define BATCH 16
#define NCLS 10
#define HW (512 * 512)
#define HW4 (HW / 4)
#define BLOCKS_PER_IMG 64
#define THREADS 256
#define GROUPS_PER_THREAD 4   // HW4 / (BLOCKS_PER_IMG * THREADS)
#define SMOOTH_F 1e-5f

typedef float v2f __attribute__((ext_vector_type(2)));
typedef float v8f __attribute__((ext_vector_type(8)));

// Stage 1: each block accumulates 32 partial sums over its pixel slice and
// reduces 256 threads -> 32 sums with chained V_WMMA_F32_16X16X4_F32
// (A = ones, B = accumulator tiles, C = running sum). Deterministic, exact f32.
__global__ __launch_bounds__(THREADS) void ce_dice_partial(
    const float* __restrict__ pred, const int* __restrict__ target,
    float* __restrict__ partials)
{
    const int tid = threadIdx.x;
    const int b   = blockIdx.y;
    const int blk = blockIdx.x;

    const float4* __restrict__ pbase =
        (const float4*)(pred + (size_t)b * NCLS * HW);
    const int4* __restrict__ tbase =
        (const int4*)(target + (size_t)b * HW);

    // acc[0..9]=masked prob sums, [10..19]=intersection, [20..29]=one-hot count,
    // [30]=nll sum, [31]=valid count  (all float so one WMMA path reduces all)
    float acc[32];
    #pragma unroll
    for (int j = 0; j < 32; ++j) acc[j] = 0.0f;

    const int f = blk * THREADS + tid;  // 0..16383 within image
    #pragma unroll
    for (int k = 0; k < GROUPS_PER_THREAD; ++k) {
        const int g = f + k * (BLOCKS_PER_IMG * THREADS);  // float4 group index
        float x[NCLS][4];
        #pragma unroll
        for (int c = 0; c < NCLS; ++c) {
            float4 v = pbase[(size_t)c * HW4 + g];   // global_load_b128, coalesced
            x[c][0] = v.x; x[c][1] = v.y; x[c][2] = v.z; x[c][3] = v.w;
        }
        const int4 tv = tbase[g];
        const int tgt[4] = {tv.x, tv.y, tv.z, tv.w};

        #pragma unroll
        for (int px = 0; px < 4; ++px) {
            const int   t  = tgt[px];
            const float vf = (t != 255) ? 1.0f : 0.0f;   // ignore-index mask

            float m = x[0][px];
            #pragma unroll
            for (int c = 1; c < NCLS; ++c) m = fmaxf(m, x[c][px]);

            float e[NCLS];
            float s = 0.0f;
            #pragma unroll
            for (int c = 0; c < NCLS; ++c) {
                e[c] = __expf(x[c][px] - m);
                s += e[c];
            }
            const float rs  = __builtin_amdgcn_rcpf(s);
            const float pv  = rs * vf;          // fold valid-mask into softmax scale
            const float lse = __logf(s) + m;    // logsumexp

            float xt = 0.0f;                    // logit of target class (gathered)
            #pragma unroll
            for (int c = 0; c < NCLS; ++c) {
                const float mc = (t == c) ? 1.0f : 0.0f;  // one-hot (255 never hits)
                const float p  = e[c] * pv;               // masked probability
                acc[c]      += p;        // union: sum p
                acc[10 + c] += p * mc;   // intersection
                acc[20 + c] += mc;       // union: sum one_hot
                xt          += mc * x[c][px];
            }
            acc[30] += (lse - xt) * vf;  // masked NLL
            acc[31] += vf;               // valid count
        }
    }

    // Stage accumulators to LDS (pad row to 33 to kill bank conflicts)
    __shared__ float smem[THREADS * 33];
    #pragma unroll
    for (int j = 0; j < 32; ++j) smem[tid * 33 + j] = acc[j];
    __syncthreads();

    // Wave 0 (all 32 lanes active -> EXEC all-ones as WMMA requires) reduces
    // the 256x32 tile with 128 chained f32 WMMAs: D[m,n] = C + sum_k B[k,n].
    if (tid < 32) {
        v8f clo = {0.f,0.f,0.f,0.f,0.f,0.f,0.f,0.f};
        v8f chi = {0.f,0.f,0.f,0.f,0.f,0.f,0.f,0.f};
        v2f a; a[0] = 1.0f; a[1] = 1.0f;        // A = ones(16x4), layout-invariant

        const int col   = tid & 15;             // component column 0..15
        const int rbase = (tid >> 4) << 1;      // lanes 16..31 carry K=2,3 rows

        for (int t0 = 0; t0 < THREADS; t0 += 4) {
            const int r0 = t0 + rbase;
            v2f blo, bhi;
            blo[0] = smem[r0 * 33 + col];            // B VGPR0: rows K=0 / K=2
            blo[1] = smem[(r0 + 1) * 33 + col];      // B VGPR1: rows K=1 / K=3
            bhi[0] = smem[r0 * 33 + col + 16];
            bhi[1] = smem[(r0 + 1) * 33 + col + 16];
            clo = __builtin_amdgcn_wmma_f32_16x16x4_f32(
                false, a, false, blo, (short)0, clo, false, false);
            chi = __builtin_amdgcn_wmma_f32_16x16x4_f32(
                false, a, false, bhi, (short)0, chi, false, false);
        }
        // All D rows are identical (A rows identical); row M=0 lives in VGPR0,
        // lanes 0..15 hold N=0..15.
        if (tid < 16) {
            float* op = partials + (size_t)(b * BLOCKS_PER_IMG + blk) * 32;
            op[tid]      = clo[0];   // components 0..15
            op[tid + 16] = chi[0];   // components 16..31
        }
    }
}

// Stage 2: single block combines the 1024 x 32 exact partials.
__global__ __launch_bounds__(256) void ce_dice_finalize(
    const float* __restrict__ partials, float* __restrict__ out)
{
    __shared__ float sdice[160];
    __shared__ float snll[256];
    __shared__ float sval[256];
    const int t = threadIdx.x;

    float nll = 0.0f, val = 0.0f;
    #pragma unroll
    for (int k = 0; k < 4; ++k) {
        const float* p = partials + (size_t)(t + 256 * k) * 32;
        nll += p[30];
        val += p[31];
    }
    snll[t] = nll;
    sval[t] = val;

    if (t < BATCH * NCLS) {
        const int b = t / NCLS;
        const int c = t % NCLS;
        float P = 0.0f, I = 0.0f, N = 0.0f;
        for (int blk = 0; blk < BLOCKS_PER_IMG; ++blk) {
            const float* p = partials + (size_t)(b * BLOCKS_PER_IMG + blk) * 32;
            P += p[c];
            I += p[10 + c];
            N += p[20 + c];
        }
        sdice[t] = (2.0f * I + SMOOTH_F) / (P + N + SMOOTH_F);
    }
    __syncthreads();

    if (t == 0) {
        float nt = 0.0f, vt = 0.0f;
        for (int i = 0; i < 256; ++i) { nt += snll[i]; vt += sval[i]; }
        float ds = 0.0f;
        for (int i = 0; i < BATCH * NCLS; ++i) ds += sdice[i];
        const float loss_ce   = nt / vt;
        const float loss_dice = 1.0f - ds / (float)(BATCH * NCLS);
        out[0] = loss_ce + loss_dice;
    }
}

extern "C" void kernel_launch(void* const* d_in, const int* in_sizes, int n_in,
                              void* d_out, int out_size, void* d_ws, size_t ws_size,
                              hipStream_t stream) {
    const float* pred   = (const float*)d_in[0];
    const int*   target = (const int*)d_in[1];
    float* partials = (float*)d_ws;   // needs 16*64*32*4 = 128 KB
    float* out      = (float*)d_out;

    dim3 grid(BLOCKS_PER_IMG, BATCH);
    ce_dice_partial<<<grid, THREADS, 0, stream>>>(pred, target, partials);
    ce_dice_finalize<<<1, 256, 0, stream>>>(partials, out);
}